// RecSysGNN_21423296873044
// MI455X (gfx1250) — compile-verified
//
#include <hip/hip_runtime.h>

#define LATENT 64
#define TILE   256   // edges per workgroup tile

typedef int v4i __attribute__((ext_vector_type(4)));
typedef __attribute__((address_space(1))) v4i* gptr_v4i;
typedef __attribute__((address_space(3))) v4i* lptr_v4i;

// -------------------------------------------------------------------
// Propagation pass: xout[to] += norm_e * xin[from], scatter via f32 atomics.
// Each block owns a 256-edge tile. Lanes 0..127 async-stage the interleaved
// (from,to) int2 tile (2 KB), lanes 128..191 async-stage the norm tile (1 KB)
// via CDNA5 global_load_async_to_lds_b128 (ASYNCcnt). Then 256 threads x 16
// iterations cover 256 edges x 16 float4-chunks: one ds_load_b64 + one
// ds_load_b32 + one global_load_b128 gather + four global_atomic_add_f32.
__global__ void __launch_bounds__(256)
k_prop(const int* __restrict__ ft, const float* __restrict__ norm,
       const float4* __restrict__ xin, float* __restrict__ xout) {
    __shared__ int   s_ft[2 * TILE];   // interleaved from,to
    __shared__ float s_norm[TILE];

    const int tid = threadIdx.x;
    const int e0  = blockIdx.x * TILE;

    if (tid < 128) {
        const int off = tid << 2;                       // 4 ints per lane
        __builtin_amdgcn_global_load_async_to_lds_b128(
            (gptr_v4i)(void*)(ft + e0 * 2 + off),
            (lptr_v4i)(void*)(s_ft + off), 0, 0);
    } else if (tid < 192) {
        const int off = (tid - 128) << 2;
        __builtin_amdgcn_global_load_async_to_lds_b128(
            (gptr_v4i)(void*)(norm + e0 + off),
            (lptr_v4i)(void*)((int*)s_norm + off), 0, 0);
    }
    __builtin_amdgcn_s_wait_asynccnt(0);
    __syncthreads();

    const int2* s_ft2 = (const int2*)s_ft;
    const int c = tid & 15;   // float4 chunk within 64-float row
    const int r = tid >> 4;   // edge sub-slot
    #pragma unroll 4
    for (int k = 0; k < 16; ++k) {
        const int slot = k * 16 + r;
        const float nrm = s_norm[slot];        // 0.0 for padded/zero-degree edges
        if (nrm != 0.0f) {
            const int2 e = s_ft2[slot];                    // ds_load_b64
            const float4 v = xin[e.x * 16 + c];            // global_load_b128
            float* dst = xout + e.y * LATENT + c * 4;
            atomicAdd(dst + 0, nrm * v.x);                 // global_atomic_add_f32
            atomicAdd(dst + 1, nrm * v.y);
            atomicAdd(dst + 2, nrm * v.z);
            atomicAdd(dst + 3, nrm * v.w);
        }
    }
}

// deg[to[e]] += 1.0f   (exact in f32 up to 2^24; E = 1.2M)
__global__ void k_degree(const long long* __restrict__ to_, float* __restrict__ deg, int E) {
    int e = blockIdx.x * blockDim.x + threadIdx.x;
    if (e < E) atomicAdd(&deg[(int)to_[e]], 1.0f);
}

__global__ void k_dinv(const float* __restrict__ deg, float* __restrict__ dinv, int n) {
    int i = blockIdx.x * blockDim.x + threadIdx.x;
    if (i < n) {
        float d = deg[i];
        dinv[i] = (d > 0.0f) ? __frsqrt_rn(d) : 0.0f;
    }
}

// Build padded edge arrays: interleaved int2 (from,to) + per-edge norm.
__global__ void k_prep(const long long* __restrict__ ef, const long long* __restrict__ et,
                       const float* __restrict__ dinv,
                       int2* __restrict__ ft, float* __restrict__ norm, int E, int Epad) {
    int e = blockIdx.x * blockDim.x + threadIdx.x;
    if (e >= Epad) return;
    if (e < E) {
        int f = (int)ef[e], t = (int)et[e];
        ft[e]   = make_int2(f, t);
        norm[e] = dinv[f] * dinv[t];
    } else {                     // padding: norm=0 => hot loop skips
        ft[e]   = make_int2(0, 0);
        norm[e] = 0.0f;
    }
}

__global__ void k_axpy4(float4* __restrict__ acc, const float4* __restrict__ x, int n4) {
    int i = blockIdx.x * blockDim.x + threadIdx.x;
    if (i < n4) {
        float4 a = acc[i], b = x[i];
        a.x += b.x; a.y += b.y; a.z += b.z; a.w += b.w;
        acc[i] = a;
    }
}

// out half1 = emb0 ; out half2 = acc * 0.25
__global__ void k_final4(float4* __restrict__ out_emb0, float4* __restrict__ out_acc,
                         const float4* __restrict__ emb, int n4) {
    int i = blockIdx.x * blockDim.x + threadIdx.x;
    if (i < n4) {
        out_emb0[i] = emb[i];
        float4 a = out_acc[i];
        a.x *= 0.25f; a.y *= 0.25f; a.z *= 0.25f; a.w *= 0.25f;
        out_acc[i] = a;
    }
}

extern "C" void kernel_launch(void* const* d_in, const int* in_sizes, int n_in,
                              void* d_out, int out_size, void* d_ws, size_t ws_size,
                              hipStream_t stream) {
    const long long* edge = (const long long*)d_in[0];  // int64 edge_index [2,E]
    const float*     emb  = (const float*)d_in[1];      // f32 [N,64]
    const int E    = in_sizes[0] / 2;
    const int NV   = in_sizes[1] / LATENT;
    const int Epad = (E + TILE - 1) & ~(TILE - 1);
    const long long* ef = edge;
    const long long* et = edge + E;

    float* out0 = (float*)d_out;                    // emb0 slot (scratch until final)
    float* acc  = out0 + (size_t)NV * LATENT;       // averaged-output slot

    float* deg  = (float*)d_ws;                     // N floats
    float* dinv = deg + NV;                         // N floats
    int*   ft   = (int*)(dinv + NV);                // 2*Epad ints, interleaved (from,to)
    float* norm = (float*)(ft + 2 * (size_t)Epad);  // Epad floats
    float* xa   = norm + Epad;                      // N*64 floats (ping)
    float* xb   = out0;                             // N*64 floats (pong, d_out half1)

    const size_t rowBytes = (size_t)NV * LATENT * sizeof(float);
    const int tpb   = 256;
    const int nElem = NV * LATENT;
    const int n4    = nElem / 4;
    const int blks4 = (n4 + tpb - 1) / tpb;
    const int tiles = Epad / TILE;

    (void)hipMemsetAsync(deg, 0, (size_t)NV * sizeof(float), stream);
    k_degree<<<(E + tpb - 1) / tpb, tpb, 0, stream>>>(et, deg, E);
    k_dinv<<<(NV + tpb - 1) / tpb, tpb, 0, stream>>>(deg, dinv, NV);
    k_prep<<<(Epad + tpb - 1) / tpb, tpb, 0, stream>>>(ef, et, dinv, (int2*)ft, norm, E, Epad);

    // acc = emb0
    (void)hipMemcpyAsync(acc, emb, rowBytes, hipMemcpyDeviceToDevice, stream);

    // layer 1: xa = P(emb); acc += xa
    (void)hipMemsetAsync(xa, 0, rowBytes, stream);
    k_prop<<<tiles, tpb, 0, stream>>>(ft, norm, (const float4*)emb, xa);
    k_axpy4<<<blks4, tpb, 0, stream>>>((float4*)acc, (const float4*)xa, n4);

    // layer 2: xb = P(xa); acc += xb
    (void)hipMemsetAsync(xb, 0, rowBytes, stream);
    k_prop<<<tiles, tpb, 0, stream>>>(ft, norm, (const float4*)xa, xb);
    k_axpy4<<<blks4, tpb, 0, stream>>>((float4*)acc, (const float4*)xb, n4);

    // layer 3: xa = P(xb); acc += xa
    (void)hipMemsetAsync(xa, 0, rowBytes, stream);
    k_prop<<<tiles, tpb, 0, stream>>>(ft, norm, (const float4*)xb, xa);
    k_axpy4<<<blks4, tpb, 0, stream>>>((float4*)acc, (const float4*)xa, n4);

    // out = [emb0, acc/4]  (overwrites the scratch use of d_out half1)
    k_final4<<<blks4, tpb, 0, stream>>>((float4*)out0, (float4*)acc, (const float4*)emb, n4);
}